// MovingMaxMinHorizontal_2267742732763
// MI455X (gfx1250) — compile-verified
//
#include <hip/hip_runtime.h>
#include <cstdint>

// Streaming moving-max then moving-min (window L=64) per [B,C] row.
// B=64, C=64, T=4096, L=64. One row per 128-thread block, 4096 blocks.
//
// Roofline: HBM traffic ~130 MB -> ~5.6 us floor @ 23.3 TB/s. The naive
// log-doubling scheme is LDS-bound (~145 B LDS/elem). This version uses
// van Herk/Gil-Werman on 32-wide chunks held in registers (~48 B LDS/elem),
// making the kernel HBM-bound. LDS arrays use a +1-float skew per 64-block
// so the lane-stride-32 scalar accesses are bank-conflict-free.
//
// CDNA5 paths: GLOBAL_LOAD_ASYNC_TO_LDS_B128 (ASYNCcnt) stages each x row
// into LDS; s_wait_asynccnt + wave32 split barriers for ordering.

#define LFILT   64
#define TLEN    4096
#define NTHR    128
#define CONCATN 4159          /* 63 + 4096 logical elements               */
#define BUFN    4224          /* phys(4158) = 4158 + 64 = 4222 < 4224     */

__device__ __forceinline__ void async_wait0() {
#if __has_builtin(__builtin_amdgcn_s_wait_asynccnt)
  __builtin_amdgcn_s_wait_asynccnt(0);
#else
  asm volatile("s_wait_asynccnt 0" ::: "memory");
#endif
}

__device__ __forceinline__ uint32_t lds_off(const void* p) {
  // Low 32 bits of a generic pointer into LDS == LDS byte offset.
  return (uint32_t)(uintptr_t)p;
}

// Skewed LDS layout: logical l -> physical l + (l >> 6). For a 32-wide block
// b (l in [32b, 32b+31]) the skew term is the constant (b >> 1).
template <bool IS_MAX>
__device__ __forceinline__ float op2(float a, float b) {
  return IS_MAX ? fmaxf(a, b) : fminf(a, b);
}

// One van Herk stage: sliding op (window 64) over the skewed concat array
// `in` (logical length 4159), thread k produces outputs t = 32k + j,
// j = 0..31, written to `out` at logical index out_off + t (skewed).
// Decomposition of window [t, t+63]:
//   [32k+j, 32k+31]  -> suffix scan of own block (registers)
//   [32k+32, 32k+63] -> full reduction of block k+1
//   [32k+64, 32k+63+j] (j>0) -> streamed prefix of block k+2
template <bool IS_MAX>
__device__ __forceinline__ void van_herk_stage(const float* in, float* out,
                                               int out_off, int k) {
  const int b0 = 32 * k       + (k >> 1);            // phys base, own block
  const int b1 = 32 * (k + 1) + ((k + 1) >> 1);      // phys base, block k+1
  const int b2 = 32 * (k + 2) + (k >> 1) + 1;        // phys base, block k+2

  float suf[32];
  suf[31] = in[b0 + 31];
#pragma unroll
  for (int j = 30; j >= 0; --j)
    suf[j] = op2<IS_MAX>(in[b0 + j], suf[j + 1]);

  float full = in[b1];
#pragma unroll
  for (int j = 1; j < 32; ++j)
    full = op2<IS_MAX>(full, in[b1 + j]);

  float p = 0.0f;
#pragma unroll
  for (int j = 0; j < 32; ++j) {
    float o = op2<IS_MAX>(suf[j], full);
    if (j > 0) {
      p = (j == 1) ? in[b2] : op2<IS_MAX>(p, in[b2 + j - 1]);
      o = op2<IS_MAX>(o, p);
    }
    const int lo = out_off + 32 * k + j;             // logical output index
    out[lo + (lo >> 6)] = o;                         // skewed store
  }
}

__global__ __launch_bounds__(NTHR)
void moving_maxmin_kernel(const float* __restrict__ x,
                          const float* __restrict__ prev_input,
                          const float* __restrict__ prev_max_out,
                          float* __restrict__ out) {
  // A: skewed concat1 (prev_input ++ x); later reused as skewed final output.
  // B: first 4096 floats double as the linear async-load staging strip for x
  //    (dead before B's first real write); then B holds skewed concat2
  //    (prev_max_out ++ max_out).
  __shared__ __attribute__((aligned(16))) float A[BUFN];
  __shared__ __attribute__((aligned(16))) float B[BUFN];

  const int tid = threadIdx.x;                       // 0..127
  const int row = blockIdx.x;                        // b*C + c, 0..4095

  const float* xrow = x            + (size_t)row * TLEN;
  const float* pin  = prev_input   + (size_t)row * (LFILT - 1);
  const float* pmo  = prev_max_out + (size_t)row * (LFILT - 1);
  float*       orow = out          + (size_t)row * TLEN;

  // ---- 1. async-load x row (16 KB) into linear staging B[0..4095] ----
  {
    uint32_t ldst = lds_off(&B[0]) + (uint32_t)tid * 16u;
    uint64_t gsrc = (uint64_t)(uintptr_t)xrow + (uint64_t)tid * 16u;
#pragma unroll
    for (int i = 0; i < 8; ++i) {                    // 8 * 128 lanes * 16B
      asm volatile("global_load_async_to_lds_b128 %0, %1, off"
                   :: "v"(ldst), "v"(gsrc) : "memory");
      ldst += (uint32_t)NTHR * 16u;
      gsrc += (uint64_t)NTHR * 16u;
    }
  }
  // prev_input -> A logical [0..62] (phys == logical there; skew term 0)
  if (tid < LFILT - 1) A[tid] = pin[tid];

  async_wait0();
  __syncthreads();

  // ---- 2. re-skew: staging B[idx] -> A at logical 63+idx ----
#pragma unroll
  for (int i = 0; i < TLEN / NTHR; ++i) {            // 32 iterations
    const int idx = i * NTHR + tid;
    const int l   = 63 + idx;
    A[l + (l >> 6)] = B[idx];
  }
  __syncthreads();

  // ---- 3. stage 1: sliding MAX over concat1 -> B at logical 63+t ----
  if (tid < LFILT - 1) B[tid] = pmo[tid];            // concat2 head
  van_herk_stage<true>(A, B, 63, tid);
  __syncthreads();

  // ---- 4. stage 2: sliding MIN over concat2 -> A at logical t ----
  van_herk_stage<false>(B, A, 0, tid);
  __syncthreads();

  // ---- 5. gather skewed result -> coalesced float4 global stores ----
#pragma unroll
  for (int i = 0; i < TLEN / (4 * NTHR); ++i) {      // 8 iterations
    const int m = i * NTHR + tid;                    // float4 index
    const int t = 4 * m;
    float4 v;
    v.x = A[(t + 0) + ((t + 0) >> 6)];
    v.y = A[(t + 1) + ((t + 1) >> 6)];
    v.z = A[(t + 2) + ((t + 2) >> 6)];
    v.w = A[(t + 3) + ((t + 3) >> 6)];
    *reinterpret_cast<float4*>(orow + t) = v;
  }
}

extern "C" void kernel_launch(void* const* d_in, const int* in_sizes, int n_in,
                              void* d_out, int out_size, void* d_ws, size_t ws_size,
                              hipStream_t stream) {
  (void)in_sizes; (void)n_in; (void)d_ws; (void)ws_size; (void)out_size;
  const float* x            = (const float*)d_in[0];   // [64,64,4096]
  const float* prev_input   = (const float*)d_in[1];   // [64,64,63]
  const float* prev_max_out = (const float*)d_in[2];   // [64,64,63]
  float*       out          = (float*)d_out;           // [64,64,4096]

  const int rows = 64 * 64;  // B*C
  moving_maxmin_kernel<<<rows, NTHR, 0, stream>>>(x, prev_input, prev_max_out, out);
}